// SSGGraphConvolution_70669391888693
// MI455X (gfx1250) — compile-verified
//
#include <hip/hip_runtime.h>

#define F 256
#define KHOPS 8
#define ALPHA_C 0.1f
#define COEF_C 0.1125f   // (1-ALPHA)/K

typedef float v2f __attribute__((ext_vector_type(2)));
typedef float v8f __attribute__((ext_vector_type(8)));

// ---------------- elementwise / setup kernels ----------------

__global__ void k_fill(float* __restrict__ p, float v, long n) {
  long i = (long)blockIdx.x * blockDim.x + threadIdx.x;
  if (i < n) p[i] = v;
}

__global__ void k_deg_scatter(const int* __restrict__ col, const float* __restrict__ w,
                              float* __restrict__ deg, int E) {
  int e = blockIdx.x * blockDim.x + threadIdx.x;
  if (e < E) atomicAdd(&deg[col[e]], w[e]);
}

__global__ void k_dinv(const float* __restrict__ deg, float* __restrict__ dinv, int n) {
  int i = blockIdx.x * blockDim.x + threadIdx.x;
  if (i < n) {
    float d = deg[i];
    dinv[i] = d > 0.0f ? rsqrtf(d) : 0.0f;
  }
}

__global__ void k_norm(const int* __restrict__ row, const int* __restrict__ col,
                       const float* __restrict__ w, const float* __restrict__ dinv,
                       float* __restrict__ norm, int E) {
  int e = blockIdx.x * blockDim.x + threadIdx.x;
  if (e < E) norm[e] = dinv[row[e]] * w[e] * dinv[col[e]];
}

// xout[i,f] = dinv[i]^2 * xin[i,f]   (self-loop term, initializes next xk)
__global__ void k_selfloop(const float* __restrict__ xin, const float* __restrict__ dinv,
                           float* __restrict__ xout, long n) {
  long i = (long)blockIdx.x * blockDim.x + threadIdx.x;
  if (i < n) {
    int node = (int)(i >> 8);  // F == 256
    float di = dinv[node];
    xout[i] = di * di * xin[i];
  }
}

// fused: hsum += xk ; xout = dinv^2 * xk   (single read of xk per hop)
__global__ void k_accum_selfloop(float* __restrict__ hsum, const float* __restrict__ xk,
                                 const float* __restrict__ dinv, float* __restrict__ xout,
                                 long n) {
  long i = (long)blockIdx.x * blockDim.x + threadIdx.x;
  if (i < n) {
    int node = (int)(i >> 8);
    float di = dinv[node];
    float v = xk[i];
    hsum[i] += v;
    xout[i] = di * di * v;
  }
}

// hsum += xk  (final hop, vectorized float4)
__global__ void k_accum(float* __restrict__ hsum, const float* __restrict__ xk, long n4) {
  long i = (long)blockIdx.x * blockDim.x + threadIdx.x;
  if (i < n4) {
    float4* h = (float4*)hsum;
    const float4* s = (const float4*)xk;
    float4 a = h[i];
    float4 b = s[i];
    a.x += b.x; a.y += b.y; a.z += b.z; a.w += b.w;
    h[i] = a;
  }
}

// 64 threads per edge, 4 features each: xout[col] += norm * xin[row]
__global__ void k_edge_scatter(const int* __restrict__ row, const int* __restrict__ col,
                               const float* __restrict__ norm,
                               const float* __restrict__ xin, float* __restrict__ xout,
                               long total) {
  long gid = (long)blockIdx.x * blockDim.x + threadIdx.x;
  if (gid >= total) return;
  int e = (int)(gid >> 6);
  int f = (int)(gid & 63) << 2;
  float nm = norm[e];
  int r = row[e], c = col[e];
  const float4 v = *(const float4*)(xin + (long)r * F + f);
  float* dst = xout + (long)c * F + f;
  atomicAdd(dst + 0, nm * v.x);
  atomicAdd(dst + 1, nm * v.y);
  atomicAdd(dst + 2, nm * v.z);
  atomicAdd(dst + 3, nm * v.w);
}

// ---------------- fp32 WMMA GEMM:  out = (ALPHA*x + COEF*hsum) @ W^T + b ----------------
// One wave32 computes a 16(M) x 64(N) tile: 4 accumulators reuse each A operand 4x.
__global__ void k_gemm_wmma(const float* __restrict__ x, const float* __restrict__ hsum,
                            const float* __restrict__ W, const float* __restrict__ bias,
                            float* __restrict__ out, int n_rows) {
  const int wavesPerBlock = blockDim.x >> 5;
  const int wave = blockIdx.x * wavesPerBlock + (threadIdx.x >> 5);
  const int lane = threadIdx.x & 31;
  const int ntn = F / 64;                 // 4 tiles of 64 output features
  const int tile_m = wave / ntn;
  const int tile_n = wave % ntn;
  if (tile_m * 16 >= n_rows) return;      // wave-uniform: EXEC stays all-1s for WMMA

  const int half = lane >> 4;             // 0: K={0,1}, 1: K={2,3}
  const int l16  = lane & 15;

  int arow = tile_m * 16 + l16;           // A-matrix M index (h row)
  const bool fullTile = (tile_m * 16 + 16 <= n_rows);  // wave-uniform
  if (arow >= n_rows) arow = n_rows - 1;  // clamp keeps EXEC full; dups masked at store

  const int bcol = tile_n * 64 + l16;     // base B-matrix N index (output feature)

  const float* __restrict__ xr  = x    + (long)arow * F;
  const float* __restrict__ hr  = hsum + (long)arow * F;
  const float* __restrict__ wr0 = W + (long)(bcol +  0) * F;  // rows of W = cols of W^T
  const float* __restrict__ wr1 = W + (long)(bcol + 16) * F;
  const float* __restrict__ wr2 = W + (long)(bcol + 32) * F;
  const float* __restrict__ wr3 = W + (long)(bcol + 48) * F;

  v8f c0 = {}, c1 = {}, c2 = {}, c3 = {};
  #pragma unroll 4
  for (int kk = 0; kk < F; kk += 4) {
    const int ka = kk + half * 2;
    v2f a;
    a.x = ALPHA_C * xr[ka]     + COEF_C * hr[ka];
    a.y = ALPHA_C * xr[ka + 1] + COEF_C * hr[ka + 1];
    v2f b0, b1, b2, b3;
    b0.x = wr0[ka]; b0.y = wr0[ka + 1];
    b1.x = wr1[ka]; b1.y = wr1[ka + 1];
    b2.x = wr2[ka]; b2.y = wr2[ka + 1];
    b3.x = wr3[ka]; b3.y = wr3[ka + 1];
    c0 = __builtin_amdgcn_wmma_f32_16x16x4_f32(false, a, false, b0, (short)0, c0, false, false);
    c1 = __builtin_amdgcn_wmma_f32_16x16x4_f32(false, a, false, b1, (short)0, c1, false, false);
    c2 = __builtin_amdgcn_wmma_f32_16x16x4_f32(false, a, false, b2, (short)0, c2, false, false);
    c3 = __builtin_amdgcn_wmma_f32_16x16x4_f32(false, a, false, b3, (short)0, c3, false, false);
  }

  const float bb0 = bias[bcol +  0];
  const float bb1 = bias[bcol + 16];
  const float bb2 = bias[bcol + 32];
  const float bb3 = bias[bcol + 48];

  // C/D layout: VGPR r holds M = r + 8*half, N = lane&15 (+ subtile offset)
  const long orow0 = (long)(tile_m * 16 + half * 8);
  if (fullTile) {
    #pragma unroll
    for (int r = 0; r < 8; ++r) {
      float* o = out + (orow0 + r) * F + bcol;
      o[ 0] = c0[r] + bb0;
      o[16] = c1[r] + bb1;
      o[32] = c2[r] + bb2;
      o[48] = c3[r] + bb3;
    }
  } else {
    #pragma unroll
    for (int r = 0; r < 8; ++r) {
      if (orow0 + r < n_rows) {
        float* o = out + (orow0 + r) * F + bcol;
        o[ 0] = c0[r] + bb0;
        o[16] = c1[r] + bb1;
        o[32] = c2[r] + bb2;
        o[48] = c3[r] + bb3;
      }
    }
  }
}

// ---------------- host-side orchestration ----------------

static inline size_t alignUp(size_t v, size_t a) { return (v + a - 1) & ~(a - 1); }

extern "C" void kernel_launch(void* const* d_in, const int* in_sizes, int n_in,
                              void* d_out, int out_size, void* d_ws, size_t ws_size,
                              hipStream_t stream) {
  const float* x    = (const float*)d_in[0];       // [N, 256]
  const int*   ei   = (const int*)d_in[1];         // [2, E] flat: [0:E]=row(src), [E:2E]=col(dst)
  const float* ew   = (const float*)d_in[2];       // [E]
  const float* W    = (const float*)d_in[3];       // [256, 256]
  const float* bias = (const float*)d_in[4];       // [256]
  float* out = (float*)d_out;

  const int N = in_sizes[0] / F;
  const int E = in_sizes[2];
  const int* row = ei;
  const int* col = ei + E;

  // workspace carve-out
  char* ws = (char*)d_ws;
  size_t off = 0;
  float* deg  = (float*)(ws + off); off = alignUp(off + (size_t)N * 4, 256);
  float* dinv = (float*)(ws + off); off = alignUp(off + (size_t)N * 4, 256);
  float* nrm  = (float*)(ws + off); off = alignUp(off + (size_t)E * 4, 256);
  float* xA   = (float*)(ws + off); off = alignUp(off + (size_t)N * F * 4, 256);
  float* xB   = (float*)(ws + off); off = alignUp(off + (size_t)N * F * 4, 256);
  float* hsum = (float*)(ws + off); off = alignUp(off + (size_t)N * F * 4, 256);
  (void)ws_size;

  const long NF = (long)N * F;
  const int TB = 256;
  auto blk = [](long n, int tb) { return (unsigned)((n + tb - 1) / tb); };

  // 1) degrees (self-loop weight 1 as init), dinv, per-edge norm
  k_fill<<<blk(N, TB), TB, 0, stream>>>(deg, 1.0f, N);
  k_deg_scatter<<<blk(E, TB), TB, 0, stream>>>(col, ew, deg, E);
  k_dinv<<<blk(N, TB), TB, 0, stream>>>(deg, dinv, N);
  k_norm<<<blk(E, TB), TB, 0, stream>>>(row, col, ew, dinv, nrm, E);

  // 2) init: xA = x, hsum = 0
  hipMemcpyAsync(xA, x, (size_t)NF * 4, hipMemcpyDeviceToDevice, stream);
  k_fill<<<blk(NF, TB), TB, 0, stream>>>(hsum, 0.0f, NF);

  // 3) K hops of normalized scatter-add propagation (feature matrix is L2-resident:
  //    102MB < 192MB L2; gathers + f32 atomics resolve at L2 atomic units)
  const long scatterTotal = (long)E * 64;  // 64 threads/edge, 4 feats each
  // hop 1
  k_selfloop<<<blk(NF, TB), TB, 0, stream>>>(xA, dinv, xB, NF);
  k_edge_scatter<<<blk(scatterTotal, TB), TB, 0, stream>>>(row, col, nrm, xA, xB, scatterTotal);
  // hops 2..K: fused accumulate + next self-loop init (xB holds latest xk)
  float* curK = xB;   // latest xk
  float* spare = xA;  // reusable buffer
  for (int k = 1; k < KHOPS; ++k) {
    k_accum_selfloop<<<blk(NF, TB), TB, 0, stream>>>(hsum, curK, dinv, spare, NF);
    k_edge_scatter<<<blk(scatterTotal, TB), TB, 0, stream>>>(row, col, nrm, curK, spare, scatterTotal);
    float* t = curK; curK = spare; spare = t;
  }
  k_accum<<<blk(NF / 4, TB), TB, 0, stream>>>(hsum, curK, NF / 4);

  // 4) fused h = ALPHA*x + COEF*hsum, out = h @ W^T + b via fp32 WMMA (16x64 tile/wave)
  const int mtiles = (N + 15) / 16;
  const long waves = (long)mtiles * (F / 64);
  const int wavesPerBlock = TB / 32;  // 8
  k_gemm_wmma<<<blk(waves, wavesPerBlock), TB, 0, stream>>>(x, hsum, W, bias, out, N);
}